// GMPOmodel_6794638262737
// MI455X (gfx1250) — compile-verified
//
#include <hip/hip_runtime.h>

// CDNA5 / gfx1250 wave32 WMMA types
typedef __attribute__((ext_vector_type(2))) float v2f;
typedef __attribute__((ext_vector_type(8))) float v8f;

constexpr int DD  = 1024;   // bond dimension
constexpr int BM  = 64;     // block tile M
constexpr int BN  = 64;     // block tile N
constexpr int BK  = 32;     // K tile staged in LDS (32 wmma between barriers)
constexpr int LDS_K = 36;   // padded row stride: 36m mod 64 distinct for m=0..15
                            // -> conflict-free b64 frag loads; mult of 4 -> aligned b128 stores

// ---------------------------------------------------------------------------
// Unified fp32 WMMA GEMM, C = op(A) * op(B), all matrices 1024x1024 row-major.
//   TA: use A^T   TB: use B^T
//   KBATCH: sum KBATCH (A_b, B_b) products into the same accumulators
// grid.z selects independent problems via sAz/sBz/sCz element strides.
// 128 threads = 4 waves (2x2), each wave owns a 32x32 patch (2x2 f32 16x16 acc).
// Double-buffered LDS: one barrier per K-tile, global loads overlap compute.
// ---------------------------------------------------------------------------
template<bool TA, bool TB, int KBATCH>
__global__ __launch_bounds__(128, 4)
void gemm1024(const float* __restrict__ Abase, long sAz, long sAb,
              const float* __restrict__ Bbase, long sBz, long sBb,
              float* __restrict__ Cbase, long sCz)
{
    __shared__ float As[2][BM * LDS_K];   // As[buf][m][k]
    __shared__ float Bs[2][BN * LDS_K];   // Bs[buf][n][k] (k-major rows)

    const int z = blockIdx.z;
    const float* A0 = Abase + (long)z * sAz;
    const float* B0 = Bbase + (long)z * sBz;
    float*       C  = Cbase + (long)z * sCz;

    const int m0  = blockIdx.y * BM;
    const int n0  = blockIdx.x * BN;
    const int tid = threadIdx.x;
    const int lane = tid & 31;
    const int wv   = tid >> 5;          // 0..3
    const int wm   = (wv >> 1) * 32;    // wave M offset
    const int wn   = (wv & 1) * 32;     // wave N offset
    const int lh   = lane >> 4;         // lane half -> K pair (ISA 16x16x4 f32 layout)
    const int ll   = lane & 15;

    v8f acc[2][2];
    #pragma unroll
    for (int i = 0; i < 2; ++i)
        #pragma unroll
        for (int j = 0; j < 2; ++j)
            acc[i][j] = {};

    constexpr int TPB   = DD / BK;          // K-tiles per batch term (32)
    constexpr int NTILE = KBATCH * TPB;

    float4 rA[4], rB[4];                    // register staging for next tile

    auto gload = [&](int tIdx) {
        const int b  = tIdx / TPB;
        const int k0 = (tIdx % TPB) * BK;
        const float* A = A0 + (long)b * sAb;
        const float* B = B0 + (long)b * sBb;
        #pragma unroll
        for (int h = 0; h < 4; ++h) {
            const int idx = tid + h * 128;
            if (!TA) {  // A[m][k], contiguous k
                const int m = idx >> 3, kq = (idx & 7) * 4;
                rA[h] = *(const float4*)&A[(long)(m0 + m) * DD + k0 + kq];
            } else {    // A^T: source contiguous along m
                const int k = idx >> 4, mq = (idx & 15) * 4;
                rA[h] = *(const float4*)&A[(long)(k0 + k) * DD + m0 + mq];
            }
            if (!TB) {  // B[k][n], contiguous n
                const int k = idx >> 4, nq = (idx & 15) * 4;
                rB[h] = *(const float4*)&B[(long)(k0 + k) * DD + n0 + nq];
            } else {    // B^T: source B[n][k], contiguous k
                const int n = idx >> 3, kq = (idx & 7) * 4;
                rB[h] = *(const float4*)&B[(long)(n0 + n) * DD + k0 + kq];
            }
        }
    };

    auto sstore = [&](int buf) {
        #pragma unroll
        for (int h = 0; h < 4; ++h) {
            const int idx = tid + h * 128;
            if (!TA) {
                const int m = idx >> 3, kq = (idx & 7) * 4;
                *(float4*)&As[buf][m * LDS_K + kq] = rA[h];
            } else {    // transpose into LDS
                const int k = idx >> 4, mq = (idx & 15) * 4;
                As[buf][(mq + 0) * LDS_K + k] = rA[h].x;
                As[buf][(mq + 1) * LDS_K + k] = rA[h].y;
                As[buf][(mq + 2) * LDS_K + k] = rA[h].z;
                As[buf][(mq + 3) * LDS_K + k] = rA[h].w;
            }
            if (!TB) {  // transpose into LDS
                const int k = idx >> 4, nq = (idx & 15) * 4;
                Bs[buf][(nq + 0) * LDS_K + k] = rB[h].x;
                Bs[buf][(nq + 1) * LDS_K + k] = rB[h].y;
                Bs[buf][(nq + 2) * LDS_K + k] = rB[h].z;
                Bs[buf][(nq + 3) * LDS_K + k] = rB[h].w;
            } else {
                const int n = idx >> 3, kq = (idx & 7) * 4;
                *(float4*)&Bs[buf][n * LDS_K + kq] = rB[h];
            }
        }
    };

    auto compute = [&](int buf) {
        #pragma unroll
        for (int kk = 0; kk < BK; kk += 4) {
            const int kb = kk + 2 * lh;
            v2f af[2], bfv[2];
            #pragma unroll
            for (int i = 0; i < 2; ++i)
                af[i] = *(const v2f*)&As[buf][(wm + i * 16 + ll) * LDS_K + kb];
            #pragma unroll
            for (int j = 0; j < 2; ++j)
                bfv[j] = *(const v2f*)&Bs[buf][(wn + j * 16 + ll) * LDS_K + kb];
            #pragma unroll
            for (int i = 0; i < 2; ++i)
                #pragma unroll
                for (int j = 0; j < 2; ++j)
                    acc[i][j] = __builtin_amdgcn_wmma_f32_16x16x4_f32(
                        false, af[i], false, bfv[j], (short)0, acc[i][j],
                        false, false);
        }
    };

    gload(0);
    sstore(0);
    __syncthreads();

    for (int tIdx = 0; tIdx < NTILE; ++tIdx) {
        const int cur = tIdx & 1;
        if (tIdx + 1 < NTILE) gload(tIdx + 1);   // overlap with compute below
        {   // speculative prefetch two tiles ahead (global_prefetch_b8)
            const int tp = (tIdx + 2 < NTILE) ? tIdx + 2 : NTILE - 1;
            const int b  = tp / TPB;
            const int k0 = (tp % TPB) * BK;
            const float* A = A0 + (long)b * sAb;
            const float* B = B0 + (long)b * sBb;
            const long aoff = TA ? ((long)(k0 + (tid >> 4)) * DD + m0 + (tid & 15) * 4)
                                 : ((long)(m0 + (tid >> 3)) * DD + k0 + (tid & 7) * 4);
            const long boff = TB ? ((long)(n0 + (tid >> 3)) * DD + k0 + (tid & 7) * 4)
                                 : ((long)(k0 + (tid >> 4)) * DD + n0 + (tid & 15) * 4);
            __builtin_prefetch(&A[aoff], 0, 1);
            __builtin_prefetch(&B[boff], 0, 1);
        }
        compute(cur);
        if (tIdx + 1 < NTILE) sstore(cur ^ 1);   // other buffer: no barrier needed first
        __syncthreads();
    }

    // epilogue: C/D layout -> VGPR v holds rows v (lanes 0-15) / v+8 (lanes 16-31)
    #pragma unroll
    for (int i = 0; i < 2; ++i) {
        #pragma unroll
        for (int j = 0; j < 2; ++j) {
            const int n = n0 + wn + j * 16 + ll;
            #pragma unroll
            for (int v = 0; v < 8; ++v) {
                const int m = m0 + wm + i * 16 + lh * 8 + v;
                C[(long)m * DD + n] = acc[i][j][v];
            }
        }
    }
}

// ---------------------------------------------------------------------------
// Deterministic reductions + elementwise helpers
// ---------------------------------------------------------------------------
__global__ void fill_kernel(float* p, float val, int n) {
    for (int i = blockIdx.x * blockDim.x + threadIdx.x; i < n; i += gridDim.x * blockDim.x)
        p[i] = val;
}

__global__ void sumsq_partial(const float* __restrict__ x, float* __restrict__ part, int n) {
    __shared__ float sm[256];
    float s = 0.f;
    for (int i = blockIdx.x * blockDim.x + threadIdx.x; i < n; i += gridDim.x * blockDim.x) {
        float v = x[i];
        s += v * v;
    }
    sm[threadIdx.x] = s;
    __syncthreads();
    for (int o = 128; o > 0; o >>= 1) {
        if ((int)threadIdx.x < o) sm[threadIdx.x] += sm[threadIdx.x + o];
        __syncthreads();
    }
    if (threadIdx.x == 0) part[blockIdx.x] = sm[0];
}

__global__ void dot_partial(const float* __restrict__ x, const float* __restrict__ y,
                            float* __restrict__ part, int n) {
    __shared__ float sm[256];
    float s = 0.f;
    for (int i = blockIdx.x * blockDim.x + threadIdx.x; i < n; i += gridDim.x * blockDim.x)
        s += x[i] * y[i];
    sm[threadIdx.x] = s;
    __syncthreads();
    for (int o = 128; o > 0; o >>= 1) {
        if ((int)threadIdx.x < o) sm[threadIdx.x] += sm[threadIdx.x + o];
        __syncthreads();
    }
    if (threadIdx.x == 0) part[blockIdx.x] = sm[0];
}

// mode 0: out = rsqrt(sum)  (normalization factor); mode 1: out = sum
__global__ void finalize_reduce(const float* __restrict__ part, int n,
                                float* __restrict__ out, int mode) {
    __shared__ float sm[256];
    float s = 0.f;
    for (int i = threadIdx.x; i < n; i += 256) s += part[i];
    sm[threadIdx.x] = s;
    __syncthreads();
    for (int o = 128; o > 0; o >>= 1) {
        if ((int)threadIdx.x < o) sm[threadIdx.x] += sm[threadIdx.x + o];
        __syncthreads();
    }
    if (threadIdx.x == 0) out[0] = (mode == 0) ? rsqrtf(sm[0]) : sm[0];
}

__global__ void scale_kernel(const float* __restrict__ src, float* __restrict__ dst,
                             const float* __restrict__ fac, int n) {
    const float s = fac[0];
    for (int i = blockIdx.x * blockDim.x + threadIdx.x; i < n; i += gridDim.x * blockDim.x)
        dst[i] = src[i] * s;
}

// E0 = sum(s16 * h) / (<l,r> * eig^2)
__global__ void final_e0(const float* __restrict__ s16, const float* __restrict__ h,
                         const float* __restrict__ eig, const float* __restrict__ lr,
                         float* __restrict__ out) {
    if (threadIdx.x == 0 && blockIdx.x == 0) {
        float acc = 0.f;
        for (int i = 0; i < 16; ++i) acc += s16[i] * h[i];
        const float e = eig[0];
        out[0] = acc / (lr[0] * e * e);
    }
}

// ---------------------------------------------------------------------------
// Host orchestration (all on `stream`; graph-capture safe: no mallocs/syncs)
// ---------------------------------------------------------------------------
extern "C" void kernel_launch(void* const* d_in, const int* in_sizes, int n_in,
                              void* d_out, int out_size, void* d_ws, size_t ws_size,
                              hipStream_t stream) {
    (void)in_sizes; (void)n_in; (void)out_size; (void)ws_size;
    const float* dA = (const float*)d_in[0];   // (2, 1024, 1024) f32
    const float* dh = (const float*)d_in[1];   // (2,2,2,2) f32
    float* out = (float*)d_out;                // scalar f32

    const size_t M = (size_t)DD * DD;          // 1M elements per matrix
    float* ws = (float*)d_ws;
    float* r    = ws;             // 1M
    float* lv   = r + M;          // 1M
    float* w    = lv + M;         // 1M
    float* t    = w + M;          // 2M (per-k intermediates)
    float* UL   = t + 2 * M;      // 2M
    float* UR   = UL + 2 * M;     // 2M
    float* Y    = UR + 2 * M;     // 4M (lower_{cd})
    float* X    = Y + 4 * M;      // 1M (upper_{ab}, reused)
    float* part = X + M;          // 1024
    float* scal = part + 1024;    // [0]=norm fac, [1]=eig, [2]=<l,r>, [3..18]=s_abcd

    const dim3 blk(128);
    const dim3 g1(DD / BN, DD / BM, 1);        // 16 x 16 blocks
    const dim3 g2(DD / BN, DD / BM, 2);
    const long Ml = (long)M;
    const int  Mi = (int)M;

    auto normalize_into = [&](float* dst) {
        sumsq_partial<<<256, 256, 0, stream>>>(w, part, Mi);
        finalize_reduce<<<1, 256, 0, stream>>>(part, 256, scal, 0);
        scale_kernel<<<1024, 256, 0, stream>>>(w, dst, scal, Mi);
    };
    auto mv_r_into_w = [&](const float* vin) {
        // t_k = A_k @ v ; w = sum_k t_k @ A_k^T
        gemm1024<false, false, 1><<<g2, blk, 0, stream>>>(dA, Ml, 0, vin, 0, 0, t, Ml);
        gemm1024<false, true,  2><<<g1, blk, 0, stream>>>(t, 0, Ml, dA, 0, Ml, w, 0);
    };
    auto mv_l_into_w = [&](const float* vin) {
        // t_k = A_k^T @ v ; w = sum_k t_k @ A_k
        gemm1024<true,  false, 1><<<g2, blk, 0, stream>>>(dA, Ml, 0, vin, 0, 0, t, Ml);
        gemm1024<false, false, 2><<<g1, blk, 0, stream>>>(t, 0, Ml, dA, 0, Ml, w, 0);
    };

    // v0 = ones / ||ones||  (norm of all-ones 1024x1024 is exactly 1024)
    fill_kernel<<<1024, 256, 0, stream>>>(r,  1.0f / 1024.0f, Mi);
    fill_kernel<<<1024, 256, 0, stream>>>(lv, 1.0f / 1024.0f, Mi);

    for (int it = 0; it < 150; ++it) { mv_r_into_w(r);  normalize_into(r);  }
    for (int it = 0; it < 150; ++it) { mv_l_into_w(lv); normalize_into(lv); }

    // eigval = <r, T r>   (r unit norm)
    mv_r_into_w(r);
    dot_partial<<<256, 256, 0, stream>>>(r, w, part, Mi);
    finalize_reduce<<<1, 256, 0, stream>>>(part, 256, scal + 1, 1);
    // <l, r>  (fold l-normalization into the final scalar; contraction is linear in l)
    dot_partial<<<256, 256, 0, stream>>>(lv, r, part, Mi);
    finalize_reduce<<<1, 256, 0, stream>>>(part, 256, scal + 2, 1);

    // UL_a = l^T @ A_a   ('aik,im->amk' as matrix (m,k))
    gemm1024<true,  false, 1><<<g2, blk, 0, stream>>>(lv, 0, 0, dA, Ml, 0, UL, Ml);
    // UR_b = A_b @ r     ('bkj,jn->bkn' as matrix (k,n))
    gemm1024<false, false, 1><<<g2, blk, 0, stream>>>(dA, Ml, 0, r, 0, 0, UR, Ml);
    // lower Y_cd = A_c @ A_d
    for (int c = 0; c < 2; ++c)
        for (int d = 0; d < 2; ++d)
            gemm1024<false, false, 1><<<g1, blk, 0, stream>>>(
                dA + (size_t)c * M, 0, 0, dA + (size_t)d * M, 0, 0,
                Y + (size_t)(c * 2 + d) * M, 0);
    // upper X_ab = UL_a @ UR_b, then s_abcd = <X_ab, Y_cd>_F
    for (int a = 0; a < 2; ++a)
        for (int b = 0; b < 2; ++b) {
            gemm1024<false, false, 1><<<g1, blk, 0, stream>>>(
                UL + (size_t)a * M, 0, 0, UR + (size_t)b * M, 0, 0, X, 0);
            for (int cd = 0; cd < 4; ++cd) {
                dot_partial<<<256, 256, 0, stream>>>(X, Y + (size_t)cd * M, part, Mi);
                finalize_reduce<<<1, 256, 0, stream>>>(part, 256,
                                                       scal + 3 + (a * 2 + b) * 4 + cd, 1);
            }
        }

    final_e0<<<1, 1, 0, stream>>>(scal + 3, dh, scal + 1, scal + 2, out);
}